// GaussianRenderer2D_28621662060896
// MI455X (gfx1250) — compile-verified
//
#include <hip/hip_runtime.h>
#include <hip/hip_bf16.h>
#include <math.h>

#define GLOBAL_AS __attribute__((address_space(1)))
#define LDS_AS    __attribute__((address_space(3)))

typedef float v2f __attribute__((ext_vector_type(2)));
typedef float v8f __attribute__((ext_vector_type(8)));
typedef int   v4i __attribute__((vector_size(16)));   // 16-byte chunk type for async DMA

static constexpr int   kH  = 256;
static constexpr int   kW  = 256;
static constexpr int   kNG = 512;
static constexpr float kL2E = 1.4426950408889634f;   // log2(e)
static constexpr float kNegBig = -1.0e30f;           // exp2 -> 0, no NaN risk

// ---------------------------------------------------------------------------
// Per-gaussian derived record: 16 floats (64 B), coords shifted by -128.
//  [0..5]  k0..k5 : E = k0*x^2 + k1*x*y + k2*y^2 + k3*x + k4*y + k5  (log2 dom.)
//  [6..7]  zero padding (A-operand K=6..7 for the second WMMA)
//  [8..11] bbox: umin, umax, vmin, vmax (shifted)
//  [12..14] clamped rgb, [15] zero
// ---------------------------------------------------------------------------
__global__ void gs2d_prep(const float* __restrict__ gp, float* __restrict__ ws) {
  int g = blockIdx.x * blockDim.x + threadIdx.x;
  if (g >= kNG) return;
  const float* p = gp + g * 9;
  float u = p[0], v = p[1];
  float sx = expf(p[2]), sy = expf(p[3]);
  float th = p[4];
  float cr  = fminf(fmaxf(p[5], 0.f), 1.f);
  float cg  = fminf(fmaxf(p[6], 0.f), 1.f);
  float cbl = fminf(fmaxf(p[7], 0.f), 1.f);
  float op = 1.f / (1.f + expf(-p[8]));
  float radius = fmaxf(sx, sy) * 3.0f;
  float umin = floorf(fminf(fmaxf(u - radius, 0.f), 255.f));
  float umax = floorf(fminf(fmaxf(u + radius, 0.f), 255.f));
  float vmin = floorf(fminf(fmaxf(v - radius, 0.f), 255.f));
  float vmax = floorf(fminf(fmaxf(v + radius, 0.f), 255.f));
  bool valid = (umax > umin) && (vmax > vmin);
  float d1 = 2.f * sx * sx + 1e-8f;
  float d2 = 2.f * sy * sy + 1e-8f;
  float ct = cosf(th), st = sinf(th);
  float i1 = 1.f / d1, i2 = 1.f / d2;
  float Ac = ct * ct * i1 + st * st * i2;
  float Bc = 2.f * ct * st * (i1 - i2);
  float Cc = st * st * i1 + ct * ct * i2;
  float us = u - 128.f, vs = v - 128.f;
  float k0 = -kL2E * Ac;
  float k1 = -kL2E * Bc;
  float k2 = -kL2E * Cc;
  float k3 =  kL2E * (2.f * Ac * us + Bc * vs);
  float k4 =  kL2E * (Bc * us + 2.f * Cc * vs);
  float k5 = log2f(op) - kL2E * (Ac * us * us + Bc * us * vs + Cc * vs * vs);
  if (!valid) k5 = kNegBig;
  float* w = ws + g * 16;
  w[0] = k0; w[1] = k1; w[2] = k2; w[3] = k3; w[4] = k4; w[5] = k5;
  w[6] = 0.f; w[7] = 0.f;
  w[8]  = umin - 128.f; w[9]  = umax - 128.f;
  w[10] = vmin - 128.f; w[11] = vmax - 128.f;
  w[12] = cr; w[13] = cg; w[14] = cbl; w[15] = 0.f;
}

__device__ inline float fast_exp2(float x) {
#if __has_builtin(__builtin_amdgcn_exp2f)
  return __builtin_amdgcn_exp2f(x);
#else
  return exp2f(x);
#endif
}

__device__ inline float swap16(float x) {   // exchange half-waves (wave32)
#if __has_builtin(__builtin_amdgcn_ds_swizzle)
  // group-of-32 swizzle, xor_mask=0x10, and_mask=0x1f  -> SWAPX16
  return __int_as_float(__builtin_amdgcn_ds_swizzle(__float_as_int(x), 0x401F));
#else
  return __shfl_xor(x, 16, 32);
#endif
}

// ---------------------------------------------------------------------------
// One wave = 16 pixels (one 16-wide row segment); block = 8 waves = 16x8 tile.
// Per 16-gaussian batch: 2x V_WMMA_F32_16X16X4_F32 evaluate the 16x16 log2
// exponents; lanes 0-15 blend gaussians 0-7 in order, lanes 16-31 gaussians
// 8-15; a SWAPX16 combines the halves in correct global order.
// ---------------------------------------------------------------------------
__global__ void __launch_bounds__(256) gs2d_render(const float* __restrict__ ws,
                                                   const float* __restrict__ bgcol,
                                                   float* __restrict__ out) {
  __shared__ float sp[kNG * 16];   // 32 KB of derived gaussian params
  const int t = threadIdx.x;

  // Stage params into LDS via CDNA5 async global->LDS DMA (ASYNCcnt path).
#if __has_builtin(__builtin_amdgcn_global_load_async_to_lds_b128)
  {
    const GLOBAL_AS char* gsrc = (const GLOBAL_AS char*)ws;
    LDS_AS char* ldst = (LDS_AS char*)sp;
#pragma unroll
    for (int k = 0; k < 8; ++k) {
      int byteoff = (t + k * 256) * 16;
      __builtin_amdgcn_global_load_async_to_lds_b128(
          (GLOBAL_AS v4i*)(gsrc + byteoff), (LDS_AS v4i*)(ldst + byteoff), 0, 0);
    }
  }
#if __has_builtin(__builtin_amdgcn_s_wait_asynccnt)
  __builtin_amdgcn_s_wait_asynccnt(0);
#else
  asm volatile("s_wait_asynccnt 0" ::: "memory");
#endif
#else
#pragma unroll
  for (int k = 0; k < 8; ++k) {
    int idx = t + k * 256;
    ((float4*)sp)[idx] = ((const float4*)ws)[idx];
  }
#endif
  __syncthreads();

  const int wave = t >> 5;
  const int lane = t & 31;
  const bool hi  = lane >= 16;
  const int px   = lane & 15;
  const int tx = blockIdx.x & 15;        // 16 tiles across
  const int ty = blockIdx.x >> 4;        // 32 tiles down
  const int x = tx * 16 + px;
  const int y = ty * 8 + wave;
  const float xs = (float)x - 128.f;
  const float yv = (float)y - 128.f;
  const float wx0 = (float)(tx * 16) - 128.f;
  const float wx1 = wx0 + 15.f;

  // B operand (pixel features), constant across batches.
  // K order: [x^2, x*y, y^2, x | y, 1, 0, 0]; VGPR0 = {lo:K0, hi:K2}, VGPR1 = {lo:K1, hi:K3}.
  v2f B1 = { hi ? yv * yv : xs * xs, hi ? xs : xs * yv };
  v2f B2 = { hi ? 0.f : yv,          hi ? 0.f : 1.f };

  float Trun = 1.f;                      // transmittance = 1 - alpha
  float accR = 0.f, accG = 0.f, accB = 0.f;

  for (int b = 0; b < 32; ++b) {
    const int gbase = b * 16;

    // Exact wave-uniform tile cull: lane l<16 tests gaussian gbase+l vs row.
    const float4 cull = *(const float4*)&sp[(gbase + px) * 16 + 8];
    bool pred = (!hi) & (cull.y >= wx0) & (cull.x <= wx1) &
                (cull.z <= yv) & (cull.w >= yv);
#if __has_builtin(__builtin_amdgcn_ballot_w32)
    if (__builtin_amdgcn_ballot_w32(pred) == 0u) continue;
#else
    (void)pred;
#endif

    const float* cf = &sp[(gbase + px) * 16];
    v8f E;
#if __has_builtin(__builtin_amdgcn_wmma_f32_16x16x4_f32)
    {
      // A operand: 16x4 f32 layout — lanes 0-15: K0/K1, lanes 16-31: K2/K3.
      const float2 a1 = *(const float2*)(cf + (hi ? 2 : 0));
      const float2 a2 = *(const float2*)(cf + (hi ? 6 : 4));
      v2f A1 = { a1.x, a1.y };
      v2f A2 = { a2.x, a2.y };
      v8f Z = {0.f, 0.f, 0.f, 0.f, 0.f, 0.f, 0.f, 0.f};
      E = __builtin_amdgcn_wmma_f32_16x16x4_f32(false, A1, false, B1,
                                                (short)0, Z, false, false);
      E = __builtin_amdgcn_wmma_f32_16x16x4_f32(false, A2, false, B2,
                                                (short)0, E, false, false);
    }
#else
#pragma unroll
    for (int r = 0; r < 8; ++r) {
      const float* c = &sp[(gbase + r + (hi ? 8 : 0)) * 16];
      E[r] = ((c[0] * xs + c[1] * yv + c[3]) * xs) +
             ((c[2] * yv + c[4]) * yv) + c[5];
    }
#endif

    // In-lane ordered blend of this half-batch (8 gaussians).
    float Tb = 1.f, bR = 0.f, bG = 0.f, bB = 0.f;
#pragma unroll
    for (int r = 0; r < 8; ++r) {
      const int gg = gbase + r + (hi ? 8 : 0);
      const float4 bx = *(const float4*)&sp[gg * 16 + 8];
      const float4 cl = *(const float4*)&sp[gg * 16 + 12];
      float gs = fast_exp2(E[r]);       // opacity already folded into exponent
      bool inb = (xs >= bx.x) & (xs <= bx.y) & (yv >= bx.z) & (yv <= bx.w);
      gs = inb ? gs : 0.f;
      float contrib = gs * Tb;
      bR = fmaf(contrib, cl.x, bR);
      bG = fmaf(contrib, cl.y, bG);
      bB = fmaf(contrib, cl.z, bB);
      Tb -= contrib;
    }

    // Combine the two half-waves in lo-then-hi (global) order.
    float To = swap16(Tb);
    float rO = swap16(bR);
    float gO = swap16(bG);
    float bO = swap16(bB);
    float Tlo = hi ? To : Tb, Thi = hi ? Tb : To;
    float Rlo = hi ? rO : bR, Rhi = hi ? bR : rO;
    float Glo = hi ? gO : bG, Ghi = hi ? bG : gO;
    float Blo = hi ? bO : bB, Bhi = hi ? bB : bO;
    float Rb = fmaf(Tlo, Rhi, Rlo);
    float Gb = fmaf(Tlo, Ghi, Glo);
    float Bb = fmaf(Tlo, Bhi, Blo);
    float Tbt = Tlo * Thi;
    accR = fmaf(Trun, Rb, accR);
    accG = fmaf(Trun, Gb, accG);
    accB = fmaf(Trun, Bb, accB);
    Trun *= Tbt;
  }

  if (!hi) {
    const int pix = y * kW + x;
    out[pix * 3 + 0] = fmaf(Trun, bgcol[0], accR);
    out[pix * 3 + 1] = fmaf(Trun, bgcol[1], accG);
    out[pix * 3 + 2] = fmaf(Trun, bgcol[2], accB);
    out[kH * kW * 3 + pix] = 1.f - Trun;   // alpha output
  }
}

extern "C" void kernel_launch(void* const* d_in, const int* in_sizes, int n_in,
                              void* d_out, int out_size, void* d_ws, size_t ws_size,
                              hipStream_t stream) {
  (void)in_sizes; (void)n_in; (void)out_size; (void)ws_size;
  const float* gp = (const float*)d_in[0];   // gaussian_params [512*9]
  const float* bg = (const float*)d_in[3];   // background_color [3]
  float* out = (float*)d_out;                // rgb [256*256*3] ++ alpha [256*256]
  float* ws  = (float*)d_ws;                 // 512*16 floats = 32 KB derived params

  gs2d_prep<<<2, 256, 0, stream>>>(gp, ws);
  gs2d_render<<<512, 256, 0, stream>>>(ws, bg, out);
}